// PagedAttentionModel_11072425689455
// MI455X (gfx1250) — compile-verified
//
#include <hip/hip_runtime.h>
#include <hip/hip_bf16.h>

// ---- problem dims ----
#define BB     8
#define HH     16
#define HKV    8
#define GG     2          // H / HKV
#define DD     128
#define HID    2048
#define VOCAB  32000
#define BSZ    16
#define MAXB   256
#define LMAX   4096

typedef __attribute__((ext_vector_type(2))) float v2f;
typedef __attribute__((ext_vector_type(8))) float v8f;

// ---- workspace layout (float offsets) ----
// xf padded to 16 rows so the 16x16 WMMA M-dim needs no masking on loads.
#define XF_OFF   0                       // [16][HID]
#define Q_OFF    (XF_OFF + 16 * HID)     // [8][HID]
#define KN_OFF   (Q_OFF  + BB * HID)     // [8][HKV*DD]
#define VN_OFF   (KN_OFF + BB * HKV*DD)  // [8][HKV*DD]
#define AO_OFF   (VN_OFF + BB * HKV*DD)  // [16][HID] padded
#define Y_OFF    (AO_OFF + 16 * HID)     // [16][HID] padded
#define WS_FLOATS (Y_OFF + 16 * HID)

// ============================================================
// Kernel 1: embedding gather + zero the pad rows of WMMA inputs
// ============================================================
__global__ void __launch_bounds__(256)
prep_kernel(const int* __restrict__ tokens, const float* __restrict__ embed,
            float* __restrict__ ws) {
    const int row = blockIdx.x;              // 0..15
    float* xf = ws + XF_OFF;
    float* ao = ws + AO_OFF;
    float* yy = ws + Y_OFF;
    for (int i = threadIdx.x; i < HID; i += 256) {
        float v = 0.0f;
        if (row < BB) v = embed[(size_t)tokens[row] * HID + i];
        xf[row * HID + i] = v;
        if (row >= BB) {                     // pad rows of later GEMM A-matrices
            ao[row * HID + i] = 0.0f;
            yy[row * HID + i] = 0.0f;
        }
    }
}

// ============================================================
// M=8(padded 16) x N x K GEMM via WMMA f32 16x16x4.
//   A: [16][K] row-major (rows 8..15 zero)   B: [K][N] row-major
//   C: [8][N], optional residual Res: [8][N]
// Each wave owns TWO adjacent 16x16 C tiles (32 columns).
// 32-bit unsigned offsets from uniform bases -> saddr+voffset loads.
// Distance-2 software pipeline: WMMA consumes fragments loaded two
// stages back, so s_wait_loadcnt can leave newer loads in flight.
// Requires K % 8 == 0 (true: K == 2048 for every call here).
// 8 waves (256 thr, wave32) per block -> 256 columns per block.
// ============================================================
__global__ void __launch_bounds__(256)
gemm8_wmma(const float* __restrict__ A, const float* __restrict__ Bw,
           const float* __restrict__ Res, float* __restrict__ C,
           int K, int N) {
    const int wave = threadIdx.x >> 5;
    const int lane = threadIdx.x & 31;
    const int n0 = (blockIdx.x * 8 + wave) * 32;    // 2 tiles per wave
    if (n0 >= N) return;                            // wave-uniform

    // ISA 7.12.2 f32 fragment layouts:
    //  A 16x4: lane = (M=lane&15), K-half = (lane>=16 ? 2 : 0); v0=K+0, v1=K+1
    //  B 4x16: lane = (N=lane&15), same K-half pattern
    const int mrow  = lane & 15;
    const int khalf = (lane >> 4) << 1;             // 0 or 2
    const int col   = lane & 15;

    const unsigned sN   = (unsigned)N;              // B row stride (floats)
    const unsigned step = 4u * sN;                  // one WMMA K-step of B

    unsigned aoff = (unsigned)(mrow * K + khalf);
    unsigned boff = (unsigned)(khalf * N + n0 + col);

    v8f acc0 = {}, acc1 = {};

    // prologue: fragments for steps 0 (slot0) and 1 (slot1)
    v2f a0, b00, b10, a1, b01, b11;
    a0    = *(const v2f*)(A + aoff);
    b00.x = Bw[boff];       b00.y = Bw[boff + sN];
    b10.x = Bw[boff + 16];  b10.y = Bw[boff + 16 + sN];
    aoff += 4; boff += step;
    a1    = *(const v2f*)(A + aoff);
    b01.x = Bw[boff];       b01.y = Bw[boff + sN];
    b11.x = Bw[boff + 16];  b11.y = Bw[boff + 16 + sN];
    aoff += 4; boff += step;

    const int steps = K >> 2;                       // even; >= 2
#pragma unroll 2
    for (int it = 0; it < (steps - 2) >> 1; ++it) {
        // ---- stage A: refill slot0 (step k+2), consume slot0 (step k) ----
        {
            v2f an, b0n, b1n;
            an    = *(const v2f*)(A + aoff);
            b0n.x = Bw[boff];       b0n.y = Bw[boff + sN];
            b1n.x = Bw[boff + 16];  b1n.y = Bw[boff + 16 + sN];
            __builtin_prefetch(Bw + boff + (size_t)8 * step, 0, 1);
            aoff += 4; boff += step;
            acc0 = __builtin_amdgcn_wmma_f32_16x16x4_f32(
                false, a0, false, b00, (short)0, acc0, false, false);
            acc1 = __builtin_amdgcn_wmma_f32_16x16x4_f32(
                false, a0, false, b10, (short)0, acc1, false, false);
            a0 = an; b00 = b0n; b10 = b1n;
        }
        // ---- stage B: refill slot1, consume slot1 ----
        {
            v2f an, b0n, b1n;
            an    = *(const v2f*)(A + aoff);
            b0n.x = Bw[boff];       b0n.y = Bw[boff + sN];
            b1n.x = Bw[boff + 16];  b1n.y = Bw[boff + 16 + sN];
            aoff += 4; boff += step;
            acc0 = __builtin_amdgcn_wmma_f32_16x16x4_f32(
                false, a1, false, b01, (short)0, acc0, false, false);
            acc1 = __builtin_amdgcn_wmma_f32_16x16x4_f32(
                false, a1, false, b11, (short)0, acc1, false, false);
            a1 = an; b01 = b0n; b11 = b1n;
        }
    }
    // drain both slots
    acc0 = __builtin_amdgcn_wmma_f32_16x16x4_f32(
        false, a0, false, b00, (short)0, acc0, false, false);
    acc1 = __builtin_amdgcn_wmma_f32_16x16x4_f32(
        false, a0, false, b10, (short)0, acc1, false, false);
    acc0 = __builtin_amdgcn_wmma_f32_16x16x4_f32(
        false, a1, false, b01, (short)0, acc0, false, false);
    acc1 = __builtin_amdgcn_wmma_f32_16x16x4_f32(
        false, a1, false, b11, (short)0, acc1, false, false);

    // C 16x16: VGPR v -> M = v + (lane>=16 ? 8 : 0), N = lane&15.
    // Only rows < 8 are real output.
    const int mbase = (lane >> 4) * 8;
#pragma unroll
    for (int v = 0; v < 8; ++v) {
        const int m = mbase + v;
        if (m < BB) {
            float v0 = acc0[v], v1 = acc1[v];
            const size_t o = (size_t)m * N + n0 + col;
            if (Res) { v0 += Res[o]; v1 += Res[o + 16]; }
            C[o]      = v0;
            C[o + 16] = v1;
        }
    }
}

// ============================================================
// Paged flash-decode attention, one block per (kv-head, batch).
//   Online softmax over CHUNK-token tiles; token l == pos uses the
//   freshly projected k_new/v_new (inputs are never mutated).
// ============================================================
#define CHUNK 1024

__global__ void __launch_bounds__(256)
attn_kernel(const float* __restrict__ qm,       // ws+Q_OFF  [8][HID]
            const float* __restrict__ kn,       // ws+KN_OFF [8][HKV*DD]
            const float* __restrict__ vn,       // ws+VN_OFF
            const float* __restrict__ k_cache,  // [NBLK*BS][HKV][DD]
            const float* __restrict__ v_cache,
            const int* __restrict__ positions,
            const int* __restrict__ block_table,
            float* __restrict__ attn_out) {     // ws+AO_OFF [16][HID]
    const int h   = blockIdx.x;                 // kv head
    const int b   = blockIdx.y;                 // batch
    const int tid = threadIdx.x;
    const int pos = positions[b];
    const int Lv  = pos + 1;
    const float scale = 0.08838834764831845f;   // 1/sqrt(128)

    __shared__ float qs[GG * DD];
    __shared__ float sc[GG * CHUNK];
    __shared__ float red[GG * 256];
    __shared__ float mrun[GG], srun[GG], mnew[GG], alpha[GG];

    for (int i = tid; i < GG * DD; i += 256)
        qs[i] = qm[(size_t)b * HID + (h * GG) * DD + i] * scale;
    if (tid < GG) { mrun[tid] = -__builtin_inff(); srun[tid] = 0.0f; }
    __syncthreads();

    const int* btab = block_table + b * MAXB;
    const int vg = tid >> 7;                    // (g,d) mapping for V pass
    const int vd = tid & 127;
    float acc = 0.0f;

    for (int base = 0; base < Lv; base += CHUNK) {
        const int cend = min(CHUNK, Lv - base);

        // ---- phase 1: raw scores for this chunk ----
        float lmax0 = -__builtin_inff(), lmax1 = -__builtin_inff();
        for (int lc = tid; lc < cend; lc += 256) {
            const int l = base + lc;
            const float* kptr;
            if (l == pos) kptr = kn + ((size_t)b * HKV + h) * DD;
            else {
                const int rowi = btab[l >> 4] * BSZ + (l & 15);
                kptr = k_cache + ((size_t)rowi * HKV + h) * DD;
            }
            float s0 = 0.0f, s1 = 0.0f;
#pragma unroll 8
            for (int d4 = 0; d4 < DD; d4 += 4) {
                float4 kv = *(const float4*)(kptr + d4);
                float4 q0 = *(const float4*)(qs + d4);
                float4 q1 = *(const float4*)(qs + DD + d4);
                s0 += kv.x * q0.x + kv.y * q0.y + kv.z * q0.z + kv.w * q0.w;
                s1 += kv.x * q1.x + kv.y * q1.y + kv.z * q1.z + kv.w * q1.w;
            }
            sc[lc] = s0; sc[CHUNK + lc] = s1;
            lmax0 = fmaxf(lmax0, s0); lmax1 = fmaxf(lmax1, s1);
        }
        red[tid] = lmax0; red[256 + tid] = lmax1;
        __syncthreads();
        for (int s = 128; s > 0; s >>= 1) {
            if (tid < s) {
                red[tid]       = fmaxf(red[tid],       red[tid + s]);
                red[256 + tid] = fmaxf(red[256 + tid], red[256 + tid + s]);
            }
            __syncthreads();
        }
        if (tid < GG) {
            const float mo = mrun[tid];
            const float mn = fmaxf(mo, red[tid * 256]);
            mnew[tid]  = mn;
            alpha[tid] = (mo == -__builtin_inff()) ? 0.0f : __expf(mo - mn);
        }
        __syncthreads();

        // ---- phase 2: exponentiate + partial sums ----
        float lsum0 = 0.0f, lsum1 = 0.0f;
        for (int lc = tid; lc < cend; lc += 256) {
            const float p0 = __expf(sc[lc]         - mnew[0]);
            const float p1 = __expf(sc[CHUNK + lc] - mnew[1]);
            sc[lc] = p0; sc[CHUNK + lc] = p1;
            lsum0 += p0; lsum1 += p1;
        }
        red[tid] = lsum0; red[256 + tid] = lsum1;
        __syncthreads();
        for (int s = 128; s > 0; s >>= 1) {
            if (tid < s) {
                red[tid]       += red[tid + s];
                red[256 + tid] += red[256 + tid + s];
            }
            __syncthreads();
        }
        if (tid < GG) {
            srun[tid] = srun[tid] * alpha[tid] + red[tid * 256];
            mrun[tid] = mnew[tid];
        }

        // ---- phase 3: weighted V accumulation (thread = (g,d)) ----
        acc *= alpha[vg];
#pragma unroll 4
        for (int lc = 0; lc < cend; ++lc) {
            const int l = base + lc;
            const float* vptr;
            if (l == pos) vptr = vn + ((size_t)b * HKV + h) * DD;
            else {
                const int rowi = btab[l >> 4] * BSZ + (l & 15);
                vptr = v_cache + ((size_t)rowi * HKV + h) * DD;
            }
            acc += sc[vg * CHUNK + lc] * vptr[vd];
        }
        __syncthreads();                        // sc/red reused next chunk
    }

    attn_out[(size_t)b * HID + (h * GG + vg) * DD + vd] = acc / srun[vg];
}

// ============================================================
// launcher
// ============================================================
extern "C" void kernel_launch(void* const* d_in, const int* in_sizes, int n_in,
                              void* d_out, int out_size, void* d_ws, size_t ws_size,
                              hipStream_t stream) {
    const int*   tokens      = (const int*)  d_in[0];
    const int*   positions   = (const int*)  d_in[1];
    const int*   block_table = (const int*)  d_in[2];
    const float* k_cache     = (const float*)d_in[3];
    const float* v_cache     = (const float*)d_in[4];
    const float* embed       = (const float*)d_in[5];
    const float* Wq          = (const float*)d_in[6];
    const float* Wk          = (const float*)d_in[7];
    const float* Wv          = (const float*)d_in[8];
    const float* Wo          = (const float*)d_in[9];
    const float* Wlm         = (const float*)d_in[10];
    float* out = (float*)d_out;
    float* ws  = (float*)d_ws;

    // 1) embed gather + pad-row zeroing
    prep_kernel<<<16, 256, 0, stream>>>(tokens, embed, ws);

    // 2) Q/K/V projections (xf[16pad][2048] @ W); 256 cols per block
    gemm8_wmma<<<HID / 256, 256, 0, stream>>>(
        ws + XF_OFF, Wq, nullptr, ws + Q_OFF, HID, HID);
    gemm8_wmma<<<(HKV * DD) / 256, 256, 0, stream>>>(
        ws + XF_OFF, Wk, nullptr, ws + KN_OFF, HID, HKV * DD);
    gemm8_wmma<<<(HKV * DD) / 256, 256, 0, stream>>>(
        ws + XF_OFF, Wv, nullptr, ws + VN_OFF, HID, HKV * DD);

    // 3) paged decode attention
    attn_kernel<<<dim3(HKV, BB), 256, 0, stream>>>(
        ws + Q_OFF, ws + KN_OFF, ws + VN_OFF, k_cache, v_cache,
        positions, block_table, ws + AO_OFF);

    // 4) o_proj + residual:  y = attn_out @ Wo + xf
    gemm8_wmma<<<HID / 256, 256, 0, stream>>>(
        ws + AO_OFF, Wo, ws + XF_OFF, ws + Y_OFF, HID, HID);

    // 5) LM head: logits = y @ Wlm   (streams 262 MB of Wlm once)
    gemm8_wmma<<<VOCAB / 256, 256, 0, stream>>>(
        ws + Y_OFF, Wlm, nullptr, out, HID, VOCAB);
}